// MultiBoxLoss_5738076307955
// MI455X (gfx1250) — compile-verified
//
#include <hip/hip_runtime.h>
#include <hip/hip_bf16.h>
#include <cstdint>
#include <math.h>

// ---------------------------------------------------------------------------
// SSD MultiBox loss for MI455X (gfx1250).
// Memory-bound (~110 MB of pred_cls/pred_loc @ 23.3 TB/s => ~5us floor).
// Strategy: 1 block per image; stream pred_cls through LDS with CDNA5
// async global->LDS DMA using a DEPTH-4 ring buffer (up to 3 tiles / 18
// async b128 ops per wave outstanding => ~14MB of DMA in flight machine-wide,
// sized to cover HBM latency at 23.3 TB/s). All matching / CE / hard-negative
// selection done on-chip; deterministic fixed-order reductions (no float
// atomics); MSB radix-select replaces the reference's full sort.
// ---------------------------------------------------------------------------

#define BATCH 128
#define NPRI  8732
#define NOBJ  16
#define NCLS  21
#define TPB   256
#define TILE  256
#define NBUF  4                                          // ring depth
#define NTILES        ((NPRI + TILE - 1) / TILE)         // 35 (last tile = 28 priors)
#define CHUNKS_FULL   ((TILE * NCLS * 4) / 16)           // 1344 x 16B chunks per tile
#define CH_PER_THREAD ((CHUNKS_FULL + TPB - 1) / TPB)    // 6

// ---- dynamic LDS layout (bytes) -------------------------------------------
#define OFF_NEGCE 0
#define SZ_NEGCE  34944                       // >= NPRI*4 (34928), >= phase-A scratch (32768)
#define OFF_CLS   (OFF_NEGCE + SZ_NEGCE)
#define SZ_CLS    (NBUF * TILE * NCLS * 4)    // 86016 (4-deep ring)
#define OFF_OBOX  (OFF_CLS + SZ_CLS)          // 16 boxes, xy form
#define OFF_OCXCY (OFF_OBOX + NOBJ * 4 * 4)   // 16 boxes, cxcywh form
#define OFF_OAREA (OFF_OCXCY + NOBJ * 4 * 4)
#define OFF_OLAB  (OFF_OAREA + NOBJ * 4)
#define OFF_BESTP (OFF_OLAB + NOBJ * 4)
#define OFF_RED   (OFF_BESTP + NOBJ * 4)      // 256 floats
#define OFF_HIST  (OFF_RED + TPB * 4)         // 256 uints (also int-reduce scratch)
#define OFF_SEL   (OFF_HIST + 256 * 4)
#define LDS_BYTES (OFF_SEL + 16)              // ~123.7 KB; 2 blocks/WGP on 320KB CDNA5

// s_wait_asynccnt takes an immediate; pick among constants (uniform branch).
#define WAIT_ASYNC_IMM(n) asm volatile("s_wait_asynccnt " #n ::: "memory")

__device__ __forceinline__ void wait_async_tiles(int tiles_outstanding) {
  if (tiles_outstanding >= 3) {
    WAIT_ASYNC_IMM(18);
  } else if (tiles_outstanding == 2) {
    WAIT_ASYNC_IMM(12);
  } else if (tiles_outstanding == 1) {
    WAIT_ASYNC_IMM(6);
  } else {
    WAIT_ASYNC_IMM(0);
  }
}

__device__ __forceinline__ float smooth_l1(float d) {
  float a = fabsf(d);
  return (a < 1.f) ? 0.5f * a * a : a - 0.5f;
}

__device__ __forceinline__ float blk_sum_f(float v, float* s, int tid) {
  s[tid] = v;
  __syncthreads();
  for (int st = TPB >> 1; st > 0; st >>= 1) {
    if (tid < st) s[tid] += s[tid + st];
    __syncthreads();
  }
  float r = s[0];
  __syncthreads();
  return r;
}

__device__ __forceinline__ int blk_sum_i(int v, int* s, int tid) {
  s[tid] = v;
  __syncthreads();
  for (int st = TPB >> 1; st > 0; st >>= 1) {
    if (tid < st) s[tid] += s[tid + st];
    __syncthreads();
  }
  int r = s[0];
  __syncthreads();
  return r;
}

// Async-DMA one pred_cls tile (nbytes, multiple of 16) into LDS at lds_base.
// Every thread issues exactly CH_PER_THREAD b128 ops (clamped duplicates for
// partial tiles) so every wave's ASYNCcnt advances by the same constant.
__device__ __forceinline__ void async_prefetch_tile(const float* gsrc, int nbytes,
                                                    uint32_t lds_base, int tid) {
  uint64_t base = (uint64_t)gsrc;
  int nchunk = nbytes >> 4;
#pragma unroll
  for (int it = 0; it < CH_PER_THREAD; ++it) {
    int c = it * TPB + tid;
    if (c >= nchunk) c = nchunk - 1;  // clamp: duplicate load, same data, stays in range
    uint32_t voff = (uint32_t)c << 4;
    uint32_t laddr = lds_base + voff;
    asm volatile("global_load_async_to_lds_b128 %0, %1, %2 offset:0"
                 :
                 : "v"(laddr), "v"(voff), "s"(base)
                 : "memory");
  }
}

__global__ void __launch_bounds__(TPB) ssd_multibox_loss_main(
    const float* __restrict__ pred_loc, const float* __restrict__ pred_cls,
    const float* __restrict__ b_boxes, const int* __restrict__ b_labels,
    const float* __restrict__ priors, float* __restrict__ ws) {
  extern __shared__ unsigned char smem[];
  float*    s_negce = (float*)(smem + OFF_NEGCE);
  float*    s_cls   = (float*)(smem + OFF_CLS);
  float*    s_obox  = (float*)(smem + OFF_OBOX);
  float*    s_ocxcy = (float*)(smem + OFF_OCXCY);
  float*    s_oarea = (float*)(smem + OFF_OAREA);
  int*      s_olab  = (int*)(smem + OFF_OLAB);
  int*      s_bestp = (int*)(smem + OFF_BESTP);
  float*    s_red   = (float*)(smem + OFF_RED);
  unsigned* s_hist  = (unsigned*)(smem + OFF_HIST);
  int*      s_sel   = (int*)(smem + OFF_SEL);

  const int b = blockIdx.x;
  const int tid = threadIdx.x;
  const float4* prv = (const float4*)priors;
  const float* cls_img = pred_cls + (size_t)b * NPRI * NCLS;
  const uint32_t cls_lds0 = (uint32_t)(size_t)(void*)s_cls;

  // Prologue: launch DMA for tiles 0..NBUF-2 immediately; the transfers
  // overlap the GT-object load and the whole of matching phase A.
#pragma unroll
  for (int pt = 0; pt < NBUF - 1; ++pt) {
    if (pt < NTILES) {
      const int pstart = pt * TILE;
      const int ppri = min(TILE, NPRI - pstart);
      async_prefetch_tile(cls_img + (size_t)pstart * NCLS, ppri * NCLS * 4,
                          cls_lds0 + (uint32_t)(pt * TILE * NCLS * 4), tid);
    }
  }

  // ---- load GT objects ----------------------------------------------------
  if (tid < NOBJ) {
    const float4 bx = ((const float4*)(b_boxes + (size_t)b * NOBJ * 4))[tid];
    s_obox[tid * 4 + 0] = bx.x;
    s_obox[tid * 4 + 1] = bx.y;
    s_obox[tid * 4 + 2] = bx.z;
    s_obox[tid * 4 + 3] = bx.w;
    float w = bx.z - bx.x, h = bx.w - bx.y;
    s_ocxcy[tid * 4 + 0] = (bx.x + bx.z) * 0.5f;
    s_ocxcy[tid * 4 + 1] = (bx.y + bx.w) * 0.5f;
    s_ocxcy[tid * 4 + 2] = w;
    s_ocxcy[tid * 4 + 3] = h;
    s_oarea[tid] = w * h;
    s_olab[tid] = b_labels[b * NOBJ + tid];
  }
  __syncthreads();

  // ---- Phase A: best prior per object (argmax over P, first-index ties) ---
  float bv[NOBJ];
  int bi[NOBJ];
#pragma unroll
  for (int o = 0; o < NOBJ; ++o) {
    bv[o] = -1.f;
    bi[o] = 0x7fffffff;
  }
  for (int p = tid; p < NPRI; p += TPB) {
    float4 pr = prv[p];
    float px1 = pr.x - pr.z * 0.5f, py1 = pr.y - pr.w * 0.5f;
    float px2 = pr.x + pr.z * 0.5f, py2 = pr.y + pr.w * 0.5f;
    float pa = pr.z * pr.w;
#pragma unroll
    for (int o = 0; o < NOBJ; ++o) {
      float iw = fminf(px2, s_obox[o * 4 + 2]) - fmaxf(px1, s_obox[o * 4 + 0]);
      float ih = fminf(py2, s_obox[o * 4 + 3]) - fmaxf(py1, s_obox[o * 4 + 1]);
      float inter = fmaxf(iw, 0.f) * fmaxf(ih, 0.f);
      float iou = inter / (pa + s_oarea[o] - inter);
      if (iou > bv[o]) {  // strict > keeps smallest p within a thread
        bv[o] = iou;
        bi[o] = p;
      }
    }
  }
  {
    // reduce over threads through the (not yet live) negce region
    float* pav = s_negce;
    int* pai = (int*)(s_negce + NOBJ * TPB);
#pragma unroll
    for (int o = 0; o < NOBJ; ++o) {
      pav[o * TPB + tid] = bv[o];
      pai[o * TPB + tid] = bi[o];
    }
    __syncthreads();
    if (tid < NOBJ) {
      float best = -1.f;
      int bidx = 0x7fffffff;
      for (int t2 = 0; t2 < TPB; ++t2) {
        float v = pav[tid * TPB + t2];
        int i = pai[tid * TPB + t2];
        if (v > best || (v == best && i < bidx)) {
          best = v;
          bidx = i;
        }
      }
      s_bestp[tid] = bidx;
    }
    __syncthreads();
  }

  // ---- Phase B: stream pred_cls tiles via async DMA ring ------------------
  float loc_acc = 0.f, conf_acc = 0.f;
  int npos_acc = 0;
  for (int t = 0; t < NTILES; ++t) {
    const int buf = t % NBUF;
    const int tstart = t * TILE;
    const int tpri = min(TILE, NPRI - tstart);

    // Top off the ring: issue DMA for tile t+NBUF-1 (overwrites buf used by
    // tile t-1, whose consumers finished at the previous closing barrier).
    const int nxt = t + NBUF - 1;
    if (nxt < NTILES) {
      const int nstart = nxt * TILE;
      const int npri = min(TILE, NPRI - nstart);
      async_prefetch_tile(cls_img + (size_t)nstart * NCLS, npri * NCLS * 4,
                          cls_lds0 + (uint32_t)((nxt % NBUF) * TILE * NCLS * 4), tid);
    }
    // Tiles issued beyond t: min(NBUF-1, NTILES-1-t). In-order async
    // completion => waiting down to 6*that guarantees tile t has landed.
    int ahead = NTILES - 1 - t;
    if (ahead > NBUF - 1) ahead = NBUF - 1;
    wait_async_tiles(ahead);
    __syncthreads();  // tile t visible to all waves

    if (tid < tpri) {
      const int p = tstart + tid;
      float4 pr = prv[p];
      float px1 = pr.x - pr.z * 0.5f, py1 = pr.y - pr.w * 0.5f;
      float px2 = pr.x + pr.z * 0.5f, py2 = pr.y + pr.w * 0.5f;
      float pa = pr.z * pr.w;
      float maxv = -1.f;
      int argo = 0;
#pragma unroll
      for (int o = 0; o < NOBJ; ++o) {
        float iw = fminf(px2, s_obox[o * 4 + 2]) - fmaxf(px1, s_obox[o * 4 + 0]);
        float ih = fminf(py2, s_obox[o * 4 + 3]) - fmaxf(py1, s_obox[o * 4 + 1]);
        float inter = fmaxf(iw, 0.f) * fmaxf(ih, 0.f);
        float iou = inter / (pa + s_oarea[o] - inter);
        if (s_bestp[o] == p) iou = 1.0f;  // forced best-prior match
        if (iou > maxv) {                 // strict > keeps smallest o on ties
          maxv = iou;
          argo = o;
        }
      }
      const bool pos = (maxv >= 0.5f);
      const int tc = pos ? s_olab[argo] : (NCLS - 1);
      const float* row = s_cls + (size_t)buf * TILE * NCLS + (size_t)tid * NCLS;
      float m = row[0];
#pragma unroll
      for (int j = 1; j < NCLS; ++j) m = fmaxf(m, row[j]);
      float se = 0.f;
#pragma unroll
      for (int j = 0; j < NCLS; ++j) se += __expf(row[j] - m);
      float ce = m + __logf(se) - row[tc];
      if (pos) {
        npos_acc += 1;
        conf_acc += ce;
        float4 pl = ((const float4*)pred_loc)[(size_t)b * NPRI + p];
        float g0 = (s_ocxcy[argo * 4 + 0] - pr.x) * 10.f / pr.z;
        float g1 = (s_ocxcy[argo * 4 + 1] - pr.y) * 10.f / pr.w;
        float g2 = __logf(s_ocxcy[argo * 4 + 2] / pr.z) * 5.f;
        float g3 = __logf(s_ocxcy[argo * 4 + 3] / pr.w) * 5.f;
        loc_acc += smooth_l1(pl.x - g0) + smooth_l1(pl.y - g1) +
                   smooth_l1(pl.z - g2) + smooth_l1(pl.w - g3);
        s_negce[p] = 0.f;  // positives contribute 0 to the negative pool
      } else {
        s_negce[p] = ce;
      }
    }
    __syncthreads();  // done reading buf before the ring refills it
  }

  // ---- deterministic block reductions -------------------------------------
  float loc_sum = blk_sum_f(loc_acc, s_red, tid);
  float conf_pos = blk_sum_f(conf_acc, s_red, tid);
  int npos = blk_sum_i(npos_acc, (int*)s_hist, tid);

  // ---- Phase C: hard-negative mining via MSB radix-select over s_negce ----
  int K = 3 * npos;
  if (K > NPRI) K = NPRI;
  float hard = 0.f;
  if (K > 0) {
    unsigned prefix = 0;
    int Kp = K;
    for (int s = 24; s >= 0; s -= 8) {
      s_hist[tid] = 0u;  // TPB == 256 bins
      __syncthreads();
      for (int p = tid; p < NPRI; p += TPB) {
        unsigned bits = __float_as_uint(s_negce[p]);  // ce >= 0 => uint order == float order
        bool match = (s == 24) || ((bits >> (s + 8)) == (prefix >> (s + 8)));
        if (match) atomicAdd(&s_hist[(bits >> s) & 0xFFu], 1u);
      }
      __syncthreads();
      if (tid == 0) {
        int cum = 0, chosen = 0, krem = Kp;
        for (int bin = 255; bin >= 0; --bin) {
          int c = (int)s_hist[bin];
          if (cum + c >= Kp) {
            chosen = bin;
            krem = Kp - cum;
            break;
          }
          cum += c;
        }
        s_sel[0] = chosen;
        s_sel[1] = krem;
      }
      __syncthreads();
      prefix |= ((unsigned)s_sel[0]) << s;
      Kp = s_sel[1];
      __syncthreads();
    }
    // prefix == bit pattern of the K-th largest value T
    float T = __uint_as_float(prefix);
    float sa = 0.f;
    int ca = 0;
    for (int p = tid; p < NPRI; p += TPB) {
      unsigned bits = __float_as_uint(s_negce[p]);
      if (bits > prefix) {
        sa += s_negce[p];
        ca += 1;
      }
    }
    sa = blk_sum_f(sa, s_red, tid);
    ca = blk_sum_i(ca, (int*)s_hist, tid);
    hard = sa + (float)(K - ca) * T;  // exact even with ties at T
  }

  if (tid == 0) {
    ws[b] = loc_sum;
    ws[BATCH + b] = conf_pos + hard;
    ws[2 * BATCH + b] = (float)npos;
  }
}

__global__ void __launch_bounds__(BATCH) ssd_multibox_loss_finalize(
    const float* __restrict__ ws, float* __restrict__ out) {
  __shared__ float s0[BATCH], s1[BATCH], s2[BATCH];
  int tid = threadIdx.x;
  s0[tid] = ws[tid];
  s1[tid] = ws[BATCH + tid];
  s2[tid] = ws[2 * BATCH + tid];
  __syncthreads();
  for (int st = BATCH >> 1; st > 0; st >>= 1) {
    if (tid < st) {
      s0[tid] += s0[tid + st];
      s1[tid] += s1[tid + st];
      s2[tid] += s2[tid + st];
    }
    __syncthreads();
  }
  if (tid == 0) {
    float np = s2[0];
    float loc = s0[0] / np;
    float conf = s1[0] / np;
    out[0] = loc + conf;  // total_loss
    out[1] = loc;         // loc_loss
    out[2] = conf;        // conf_loss
  }
}

extern "C" void kernel_launch(void* const* d_in, const int* in_sizes, int n_in,
                              void* d_out, int out_size, void* d_ws, size_t ws_size,
                              hipStream_t stream) {
  const float* pred_loc = (const float*)d_in[0];
  const float* pred_cls = (const float*)d_in[1];
  const float* b_boxes = (const float*)d_in[2];
  const int* b_labels = (const int*)d_in[3];
  const float* priors = (const float*)d_in[4];
  float* ws = (float*)d_ws;

  ssd_multibox_loss_main<<<dim3(BATCH), dim3(TPB), LDS_BYTES, stream>>>(
      pred_loc, pred_cls, b_boxes, b_labels, priors, ws);
  ssd_multibox_loss_finalize<<<dim3(1), dim3(BATCH), 0, stream>>>(ws, (float*)d_out);

  (void)in_sizes;
  (void)n_in;
  (void)out_size;
  (void)ws_size;
}